// GCN_75101798138154
// MI455X (gfx1250) — compile-verified
//
#include <hip/hip_runtime.h>

typedef __attribute__((ext_vector_type(2))) float v2f;
typedef __attribute__((ext_vector_type(8))) float v8f;

#define WPB 2  // waves per block in the MLP kernel

__device__ __forceinline__ void atomic_add_f32(float* p, float v) {
    __hip_atomic_fetch_add(p, v, __ATOMIC_RELAXED, __HIP_MEMORY_SCOPE_AGENT);
}

// -------- fused item MLP: temp = leaky(F @ W0 + b0, 0.01) @ W1 + b1 --------
// One wave32 per 16-row item tile, fp32 WMMA 16x16x4 throughout.
__global__ __launch_bounds__(WPB * 32) void mlp_kernel(
    const float* __restrict__ F,   // [num_item, 128]
    const float* __restrict__ W0,  // [128, 256]
    const float* __restrict__ b0,  // [256]
    const float* __restrict__ W1,  // [256, 64]
    const float* __restrict__ b1,  // [64]
    float* __restrict__ xout,      // x buffer; rows written at (num_user + row)
    int num_item, int num_user)
{
    __shared__ float hid[WPB][16][258];  // 16x256 hidden tile, padded stride

    const int wave = threadIdx.x >> 5;
    const int lane = threadIdx.x & 31;
    const int half = lane >> 4;       // 0: K/M-low half of wave, 1: high half
    const int l    = lane & 15;
    const int tile = blockIdx.x * WPB + wave;  // wave-uniform
    if (tile * 16 >= num_item) return;         // uniform exit, EXEC stays full
    const int row0 = tile * 16;

    // ---- GEMM1: hid = leaky(F[row0:row0+16, :128] @ W0 + b0) ----
    const float* frow = F + (size_t)(row0 + l) * 128 + 2 * half;
    for (int nt = 0; nt < 16; ++nt) {          // 16 col-tiles of 16 (HID=256)
        const float bv = b0[nt * 16 + l];
        v8f acc;
        #pragma unroll
        for (int r = 0; r < 8; ++r) acc[r] = bv;
        const float* wcol = W0 + (size_t)(2 * half) * 256 + nt * 16 + l;
        #pragma unroll 4
        for (int k = 0; k < 32; ++k) {         // K = 128 / 4
            v2f a = *(const v2f*)(frow + 4 * k);
            v2f b;
            b.x = wcol[(size_t)(4 * k)     * 256];
            b.y = wcol[(size_t)(4 * k + 1) * 256];
            acc = __builtin_amdgcn_wmma_f32_16x16x4_f32(
                false, a, false, b, (short)0, acc, false, false);
        }
        #pragma unroll
        for (int r = 0; r < 8; ++r) {
            float v = acc[r];
            hid[wave][r + 8 * half][nt * 16 + l] = (v > 0.0f) ? v : 0.01f * v;
        }
    }

    // ---- GEMM2: xout = hid @ W1 + b1 ----
    for (int nt = 0; nt < 4; ++nt) {           // 4 col-tiles of 16 (DLAT=64)
        const float bv = b1[nt * 16 + l];
        v8f acc;
        #pragma unroll
        for (int r = 0; r < 8; ++r) acc[r] = bv;
        #pragma unroll 4
        for (int k = 0; k < 64; ++k) {         // K = 256 / 4
            v2f a = *(const v2f*)&hid[wave][l][4 * k + 2 * half];
            v2f b;
            b.x = W1[(size_t)(4 * k + 2 * half)     * 64 + nt * 16 + l];
            b.y = W1[(size_t)(4 * k + 2 * half + 1) * 64 + nt * 16 + l];
            acc = __builtin_amdgcn_wmma_f32_16x16x4_f32(
                false, a, false, b, (short)0, acc, false, false);
        }
        #pragma unroll
        for (int r = 0; r < 8; ++r)
            xout[(size_t)(num_user + row0 + r + 8 * half) * 64 + nt * 16 + l] = acc[r];
    }
}

// -------- L2 row-normalize: one wave32 per 64-wide row --------
__global__ void normalize_kernel(float* __restrict__ x, int nrows) {
    const int w    = (int)((blockIdx.x * blockDim.x + threadIdx.x) >> 5);
    const int lane = threadIdx.x & 31;
    if (w >= nrows) return;
    float2 v = *(const float2*)&x[(size_t)w * 64 + lane * 2];
    float ss = v.x * v.x + v.y * v.y;
    #pragma unroll
    for (int m = 16; m > 0; m >>= 1) ss += __shfl_xor(ss, m, 32);
    const float s = 1.0f / fmaxf(sqrtf(ss), 1e-12f);
    v.x *= s; v.y *= s;
    *(float2*)&x[(size_t)w * 64 + lane * 2] = v;
}

// -------- degree histogram over source rows --------
__global__ void degree_kernel(const int* __restrict__ rows,
                              float* __restrict__ deg, int E) {
    int i = blockIdx.x * blockDim.x + threadIdx.x;
    const int stride = gridDim.x * blockDim.x;
    for (; i < E; i += stride) atomic_add_f32(&deg[rows[i]], 1.0f);
}

__global__ void invsqrt_kernel(float* __restrict__ d, int n) {
    const int i = blockIdx.x * blockDim.x + threadIdx.x;
    if (i < n) d[i] = 1.0f / sqrtf(d[i]);   // deg >= 1 by construction (spine)
}

// -------- GCN conv: out[col] += dis[row]*dis[col] * xin[row] --------
// Each wave owns a 32-edge chunk: lane i loads edge metadata for edge base+i
// (coalesced), then the wave iterates the 32 edges broadcasting (r,c,nv) via
// shuffles while all 32 lanes move the 64-float row (float2/lane, b64 gather +
// two global_atomic_add_f32). Keeps many independent gathers/atomics in flight
// instead of one dependent chain per edge.
__global__ void conv_kernel(const float* __restrict__ xin,
                            float* __restrict__ xout,
                            const int* __restrict__ rows,
                            const int* __restrict__ cols,
                            const float* __restrict__ dis, int E) {
    const int lane   = threadIdx.x & 31;
    const int wave   = (int)((blockIdx.x * blockDim.x + threadIdx.x) >> 5);
    const int nwaves = (int)((gridDim.x * blockDim.x) >> 5);
    for (int base = wave * 32; base < E; base += nwaves * 32) {
        const int e = base + lane;
        int r = 0, c = 0;
        float nv = 0.0f;
        if (e < E) {
            r  = rows[e];
            c  = cols[e];
            nv = dis[r] * dis[c];
        }
        int cnt = E - base;
        if (cnt > 32) cnt = 32;
        for (int j = 0; j < cnt; ++j) {
            const int   rj  = __shfl(r,  j, 32);
            const int   cj  = __shfl(c,  j, 32);
            const float nvj = __shfl(nv, j, 32);
            const float2 v = *(const float2*)&xin[(size_t)rj * 64 + lane * 2];
            float* dst = &xout[(size_t)cj * 64 + lane * 2];
            atomic_add_f32(dst,     nvj * v.x);
            atomic_add_f32(dst + 1, nvj * v.y);
        }
    }
}

// -------- x_hat = h + x + h1 (float4) --------
__global__ void final_kernel(const float4* __restrict__ h,
                             const float4* __restrict__ x,
                             const float4* __restrict__ h1,
                             float4* __restrict__ out, long n4) {
    long i = blockIdx.x * (long)blockDim.x + threadIdx.x;
    const long stride = gridDim.x * (long)blockDim.x;
    for (; i < n4; i += stride) {
        float4 a = h[i], b = x[i], c = h1[i];
        out[i] = make_float4(a.x + b.x + c.x, a.y + b.y + c.y,
                             a.z + b.z + c.z, a.w + b.w + c.w);
    }
}

extern "C" void kernel_launch(void* const* d_in, const int* in_sizes, int n_in,
                              void* d_out, int out_size, void* d_ws, size_t ws_size,
                              hipStream_t stream) {
    const int*   edges      = (const int*)d_in[0];
    const float* features   = (const float*)d_in[1];
    const float* preference = (const float*)d_in[2];
    const float* W0         = (const float*)d_in[3];
    const float* b0         = (const float*)d_in[4];
    const float* W1         = (const float*)d_in[5];
    const float* b1         = (const float*)d_in[6];
    float*       out        = (float*)d_out;

    const int E        = in_sizes[0] / 2;
    const int num_item = in_sizes[1] / 128;
    const int num_user = in_sizes[2] / 64;
    const int N        = num_user + num_item;
    const int* rows = edges;
    const int* cols = edges + E;

    // workspace carve-out: x | h | h1 | dis
    float* x   = (float*)d_ws;
    float* h   = x  + (size_t)N * 64;
    float* h1  = h  + (size_t)N * 64;
    float* dis = h1 + (size_t)N * 64;

    hipMemsetAsync(h,   0, (size_t)N * 64 * sizeof(float), stream);
    hipMemsetAsync(h1,  0, (size_t)N * 64 * sizeof(float), stream);
    hipMemsetAsync(dis, 0, (size_t)N * sizeof(float), stream);
    hipMemcpyAsync(x, preference, (size_t)num_user * 64 * sizeof(float),
                   hipMemcpyDeviceToDevice, stream);

    // item MLP -> x[num_user:]
    const int ntiles  = (num_item + 15) / 16;
    const int nblocks = (ntiles + WPB - 1) / WPB;
    mlp_kernel<<<nblocks, WPB * 32, 0, stream>>>(features, W0, b0, W1, b1, x,
                                                 num_item, num_user);

    // row normalize (8 waves / 256-thread block)
    normalize_kernel<<<(N + 7) / 8, 256, 0, stream>>>(x, N);

    // symmetric degree norm
    degree_kernel<<<1024, 256, 0, stream>>>(rows, dis, E);
    invsqrt_kernel<<<(N + 255) / 256, 256, 0, stream>>>(dis, N);

    // two scatter convs (L2-resident atomics)
    conv_kernel<<<4096, 256, 0, stream>>>(x, h, rows, cols, dis, E);
    conv_kernel<<<4096, 256, 0, stream>>>(h, h1, rows, cols, dis, E);

    // x_hat = h + x + h1 ; then append raw preference
    const long n4 = (long)N * 16;
    final_kernel<<<4096, 256, 0, stream>>>((const float4*)h, (const float4*)x,
                                           (const float4*)h1, (float4*)out, n4);
    hipMemcpyAsync(out + (size_t)N * 64, preference,
                   (size_t)num_user * 64 * sizeof(float),
                   hipMemcpyDeviceToDevice, stream);
}